// net_LSTM_21827023798542
// MI455X (gfx1250) — compile-verified
//
#include <hip/hip_runtime.h>

#define T_ 2048
#define B_ 1024
#define I_ 15
#define H_ 10
#define G_ 40   // 4*H

typedef float v2f __attribute__((ext_vector_type(2)));
typedef float v8f __attribute__((ext_vector_type(8)));

// ---------------------------------------------------------------------------
// Phase 1: pre[m][g] = sum_k x[m][k] * W_ih[g][k]   (m in [0,T*B), g in [0,40))
// f32 WMMA 16x16x4: one 16-row M tile per wave, N padded 40->48, K padded 15->16.
// ---------------------------------------------------------------------------
__global__ __launch_bounds__(256) void lstm_pre_wmma(const float* __restrict__ x,
                                                     const float* __restrict__ w_ih,
                                                     float* __restrict__ pre) {
  const int lane  = threadIdx.x & 31;
  const int wave  = threadIdx.x >> 5;
  const int tile  = blockIdx.x * 8 + wave;
  const int m0    = tile * 16;
  const int row   = lane & 15;          // M (for A) / N (for B,C) within tile
  const int half  = lane >> 4;          // 0: K pair {0,1}, 1: K pair {2,3}

  // B fragments: Bmat[k][col] = W_ih[col][k], zero-padded. Loaded once (L2-hot).
  v2f bfrag[3][4];
#pragma unroll
  for (int n = 0; n < 3; ++n) {
    const int col = n * 16 + row;
#pragma unroll
    for (int kk = 0; kk < 4; ++kk) {
#pragma unroll
      for (int v = 0; v < 2; ++v) {
        const int k = kk * 4 + half * 2 + v;
        float val = 0.0f;
        if (col < G_ && k < I_) val = w_ih[col * I_ + k];
        bfrag[n][kk][v] = val;
      }
    }
  }

  // A fragments: A[m][k] = x[m0+row][k], zero-padded along K.
  v2f afrag[4];
  const float* xr = x + (size_t)(m0 + row) * I_;
#pragma unroll
  for (int kk = 0; kk < 4; ++kk) {
#pragma unroll
    for (int v = 0; v < 2; ++v) {
      const int k = kk * 4 + half * 2 + v;
      afrag[kk][v] = (k < I_) ? xr[k] : 0.0f;
    }
  }

  v8f acc[3];
#pragma unroll
  for (int n = 0; n < 3; ++n) acc[n] = (v8f){0.f,0.f,0.f,0.f,0.f,0.f,0.f,0.f};

#pragma unroll
  for (int kk = 0; kk < 4; ++kk) {
#pragma unroll
    for (int n = 0; n < 3; ++n) {
      acc[n] = __builtin_amdgcn_wmma_f32_16x16x4_f32(
          /*neg_a=*/false, afrag[kk], /*neg_b=*/false, bfrag[n][kk],
          /*c_mod=*/(short)0, acc[n], /*reuse_a=*/false, /*reuse_b=*/false);
    }
  }

  // Store: element (M = v + 8*half, N = lane&15) per C/D layout.
#pragma unroll
  for (int n = 0; n < 3; ++n) {
    const int col = n * 16 + row;
    if (col < G_) {
#pragma unroll
      for (int v = 0; v < 8; ++v) {
        const int r = m0 + v + half * 8;
        pre[(size_t)r * G_ + col] = acc[n][v];
      }
    }
  }
}

// ---------------------------------------------------------------------------
// Phase 2: sequential scan. 4 lanes per batch element (one gate class each),
// W_hh slice in registers, gate exchange via intra-wave LDS.
// ---------------------------------------------------------------------------
__device__ __forceinline__ float sigm_(float v) {
  return 1.0f / (1.0f + __expf(-v));
}
__device__ __forceinline__ float tanh_(float v) {
  return 2.0f / (1.0f + __expf(-2.0f * v)) - 1.0f;
}

template <bool USE_PRE>
__global__ __launch_bounds__(256) void lstm_scan(const float* __restrict__ pre,
                                                 const float* __restrict__ x,
                                                 const float* __restrict__ w_ih,
                                                 const float* __restrict__ w_hh,
                                                 const float* __restrict__ h0,
                                                 const float* __restrict__ c0,
                                                 float* __restrict__ out,
                                                 float* __restrict__ hN,
                                                 float* __restrict__ cN) {
  __shared__ __align__(16) float s_act[64][4][12];          // padded rows (48B)
  __shared__ float s_wih[USE_PRE ? 1 : G_ * I_];

  const int tid = threadIdx.x;
  const int r   = tid & 3;        // gate class: 0=i 1=f 2=g 3=o
  const int bl  = tid >> 2;       // local batch index 0..63
  const int b   = blockIdx.x * 64 + bl;

  if (!USE_PRE) {
    for (int i = tid; i < G_ * I_; i += 256) s_wih[i] = w_ih[i];
    __syncthreads();
  }

  // W_hh rows [10r, 10r+10) in registers: 100 VGPRs.
  float w[H_][H_];
#pragma unroll
  for (int j = 0; j < H_; ++j)
#pragma unroll
    for (int k = 0; k < H_; ++k)
      w[j][k] = w_hh[(r * H_ + j) * H_ + k];

  float h[H_], c[H_];
#pragma unroll
  for (int j = 0; j < H_; ++j) { h[j] = h0[b * H_ + j]; c[j] = c0[b * H_ + j]; }

  for (int t = 0; t < T_; ++t) {
    const size_t rowoff = (size_t)t * B_ + b;
    float acc[H_];

    if (USE_PRE) {
      const float2* p2 = (const float2*)(pre + rowoff * G_ + r * H_);
#pragma unroll
      for (int q = 0; q < 5; ++q) { float2 v = p2[q]; acc[2*q] = v.x; acc[2*q+1] = v.y; }
      if (t + 1 < T_)
        __builtin_prefetch(pre + (rowoff + B_) * G_ + r * H_, 0, 1);
    } else {
      float xv[I_];
      const float* xp = x + rowoff * I_;
#pragma unroll
      for (int k = 0; k < I_; ++k) xv[k] = xp[k];
#pragma unroll
      for (int j = 0; j < H_; ++j) {
        float a = 0.0f;
#pragma unroll
        for (int k = 0; k < I_; ++k) a = fmaf(xv[k], s_wih[(r * H_ + j) * I_ + k], a);
        acc[j] = a;
      }
    }

    // gates += h @ W_hh[10r:10r+10]^T  (100 FMAs, all operands in VGPRs)
#pragma unroll
    for (int k = 0; k < H_; ++k)
#pragma unroll
      for (int j = 0; j < H_; ++j)
        acc[j] = fmaf(h[k], w[j][k], acc[j]);

    // activation for my gate class only (g-class uses tanh)
    float act[H_];
#pragma unroll
    for (int j = 0; j < H_; ++j)
      act[j] = (r == 2) ? tanh_(acc[j]) : sigm_(acc[j]);

    // publish to the other 3 lanes of this group (same wave -> in-order DS)
    *(float4*)&s_act[bl][r][0] = make_float4(act[0], act[1], act[2], act[3]);
    *(float4*)&s_act[bl][r][4] = make_float4(act[4], act[5], act[6], act[7]);
    *(float2*)&s_act[bl][r][8] = make_float2(act[8], act[9]);
    __builtin_amdgcn_wave_barrier();

    float iv[H_], fv[H_], gv[H_], ov[H_];
    {
      float4 a0; float2 a1;
      a0 = *(const float4*)&s_act[bl][0][0]; iv[0]=a0.x; iv[1]=a0.y; iv[2]=a0.z; iv[3]=a0.w;
      a0 = *(const float4*)&s_act[bl][0][4]; iv[4]=a0.x; iv[5]=a0.y; iv[6]=a0.z; iv[7]=a0.w;
      a1 = *(const float2*)&s_act[bl][0][8]; iv[8]=a1.x; iv[9]=a1.y;
      a0 = *(const float4*)&s_act[bl][1][0]; fv[0]=a0.x; fv[1]=a0.y; fv[2]=a0.z; fv[3]=a0.w;
      a0 = *(const float4*)&s_act[bl][1][4]; fv[4]=a0.x; fv[5]=a0.y; fv[6]=a0.z; fv[7]=a0.w;
      a1 = *(const float2*)&s_act[bl][1][8]; fv[8]=a1.x; fv[9]=a1.y;
      a0 = *(const float4*)&s_act[bl][2][0]; gv[0]=a0.x; gv[1]=a0.y; gv[2]=a0.z; gv[3]=a0.w;
      a0 = *(const float4*)&s_act[bl][2][4]; gv[4]=a0.x; gv[5]=a0.y; gv[6]=a0.z; gv[7]=a0.w;
      a1 = *(const float2*)&s_act[bl][2][8]; gv[8]=a1.x; gv[9]=a1.y;
      a0 = *(const float4*)&s_act[bl][3][0]; ov[0]=a0.x; ov[1]=a0.y; ov[2]=a0.z; ov[3]=a0.w;
      a0 = *(const float4*)&s_act[bl][3][4]; ov[4]=a0.x; ov[5]=a0.y; ov[6]=a0.z; ov[7]=a0.w;
      a1 = *(const float2*)&s_act[bl][3][8]; ov[8]=a1.x; ov[9]=a1.y;
    }
    __builtin_amdgcn_wave_barrier();

    // cell/hidden update (replicated identically across the 4 lanes)
#pragma unroll
    for (int j = 0; j < H_; ++j) {
      const float cn = fmaf(fv[j], c[j], iv[j] * gv[j]);
      c[j] = cn;
      h[j] = ov[j] * tanh_(cn);
    }

    if (r == 0) {
      float2* op = (float2*)(out + rowoff * H_);
      op[0] = make_float2(h[0], h[1]);
      op[1] = make_float2(h[2], h[3]);
      op[2] = make_float2(h[4], h[5]);
      op[3] = make_float2(h[6], h[7]);
      op[4] = make_float2(h[8], h[9]);
    }
  }

  if (r == 0) {
#pragma unroll
    for (int j = 0; j < H_; ++j) {
      hN[b * H_ + j] = h[j];
      cN[b * H_ + j] = c[j];
    }
  }
}

// ---------------------------------------------------------------------------
extern "C" void kernel_launch(void* const* d_in, const int* in_sizes, int n_in,
                              void* d_out, int out_size, void* d_ws, size_t ws_size,
                              hipStream_t stream) {
  const float* x   = (const float*)d_in[0];
  const float* h0  = (const float*)d_in[1];
  const float* c0  = (const float*)d_in[2];
  const float* wih = (const float*)d_in[3];
  const float* whh = (const float*)d_in[4];

  float* out = (float*)d_out;
  float* hN  = out + (size_t)T_ * B_ * H_;
  float* cN  = hN + (size_t)B_ * H_;

  const size_t pre_bytes = (size_t)T_ * B_ * G_ * sizeof(float);
  const int scan_blocks = (B_ * 4) / 256;  // 16

  if (ws_size >= pre_bytes) {
    float* pre = (float*)d_ws;
    const int tiles  = (T_ * B_) / 16;     // 131072
    const int blocks = tiles / 8;          // 8 waves (tiles) per 256-thread block
    lstm_pre_wmma<<<blocks, 256, 0, stream>>>(x, wih, pre);
    lstm_scan<true><<<scan_blocks, 256, 0, stream>>>(pre, x, wih, whh, h0, c0, out, hN, cN);
  } else {
    lstm_scan<false><<<scan_blocks, 256, 0, stream>>>(nullptr, x, wih, whh, h0, c0, out, hN, cN);
  }
}